// CostSensitiveLoss_76209899700359
// MI455X (gfx1250) — compile-verified
//
#include <hip/hip_runtime.h>
#include <hip/hip_bf16.h>

typedef float f32x4 __attribute__((ext_vector_type(4)));
typedef float v2f   __attribute__((ext_vector_type(2)));
typedef float v8f   __attribute__((ext_vector_type(8)));

#define NROWS 262144
#define NCOLS 1000
#define BLOCKS 1024
#define TPB 256
#define WAVES_PER_BLOCK (TPB / 32)
#define TOTAL_WAVES (BLOCKS * WAVES_PER_BLOCK) /* 8192 waves, 32 rows each */

// ---------------------------------------------------------------------------
// Kernel 1: one wave32 per row. Streaming non-temporal b128 loads, in-register
// online softmax stats, deterministic per-block partial sums into d_ws.
// ---------------------------------------------------------------------------
__global__ __launch_bounds__(TPB) void cs_loss_main(
    const float* __restrict__ outputs,
    const int*   __restrict__ labels,
    const float* __restrict__ cost,
    float*       __restrict__ partials /* 2*BLOCKS floats */)
{
    const int lane         = threadIdx.x & 31;
    const int wid_in_block = threadIdx.x >> 5;
    const int wid          = blockIdx.x * WAVES_PER_BLOCK + wid_in_block;

    const float ninf = -__builtin_inff();

    float ce_acc = 0.0f, cost_acc = 0.0f;

    for (int row = wid; row < NROWS; row += TOTAL_WAVES) {
        const float* rp    = outputs + (size_t)row * NCOLS;
        const int    label = labels[row];

        // 250 float4 per row; lane handles float4 index j = lane + 32*t
        f32x4 v[8];
#pragma unroll
        for (int t = 0; t < 8; ++t) {
            const int j = lane + 32 * t;
            if (j < 250) {
                v[t] = __builtin_nontemporal_load((const f32x4*)(rp + 4 * j));
            } else {
                v[t] = f32x4{ninf, ninf, ninf, ninf};
            }
        }

        // Lane-local max / first-occurrence argmax + label-value gather
        float bm = ninf;
        int   bi = 0x7FFFFFFF;
        float xl = 0.0f;
#pragma unroll
        for (int t = 0; t < 8; ++t) {
            const int base = 4 * (lane + 32 * t);
#pragma unroll
            for (int c2 = 0; c2 < 4; ++c2) {
                const float x   = v[t][c2];
                const int   col = base + c2;
                if (x > bm) { bm = x; bi = col; }
                if (col == label) xl = x;
            }
        }

        // Wave butterfly reduce: max with smallest-index tiebreak
#pragma unroll
        for (int off = 16; off > 0; off >>= 1) {
            const float om = __shfl_xor(bm, off, 32);
            const int   oi = __shfl_xor(bi, off, 32);
            if (om > bm || (om == bm && oi < bi)) { bm = om; bi = oi; }
        }

        // Sum of exp(x - max); padded -inf entries contribute exp(-inf) = 0
        float s = 0.0f;
#pragma unroll
        for (int t = 0; t < 8; ++t)
#pragma unroll
            for (int c2 = 0; c2 < 4; ++c2)
                s += __expf(v[t][c2] - bm);
#pragma unroll
        for (int off = 16; off > 0; off >>= 1)
            s += __shfl_xor(s, off, 32);

        // Exactly one lane holds the label's logit; reduce-add broadcasts it
#pragma unroll
        for (int off = 16; off > 0; off >>= 1)
            xl += __shfl_xor(xl, off, 32);

        if (lane == 0) {
            const float logp = (xl - bm) - __logf(s);
            ce_acc   += logp;                       // sum of logp[label]
            cost_acc += cost[label * NCOLS + bi];   // bi = argmax pred (L2-resident)
        }
    }

    // Deterministic block reduction (no atomics)
    __shared__ float red[WAVES_PER_BLOCK][2];
    if (lane == 0) {
        red[wid_in_block][0] = ce_acc;
        red[wid_in_block][1] = cost_acc;
    }
    __syncthreads();
    if (threadIdx.x == 0) {
        float ce = 0.0f, co = 0.0f;
#pragma unroll
        for (int w = 0; w < WAVES_PER_BLOCK; ++w) {
            ce += red[w][0];
            co += red[w][1];
        }
        partials[2 * blockIdx.x + 0] = ce;
        partials[2 * blockIdx.x + 1] = co;
    }
}

// ---------------------------------------------------------------------------
// Kernel 2: single wave32 reduces 1024 (ce, cost) partials.
// Uses V_WMMA_F32_16X16X4_F32 with A = ones as a matrix-pipe reduction:
// D[m][n] = colsum_n(B), constant in m, so a full-wave butterfly over D's
// first accumulator VGPR yields 2 * (sum of all 64 B slots), independent of
// the exact K/N lane mapping. EXEC is all-ones here (32 threads, no branches).
// ---------------------------------------------------------------------------
__global__ __launch_bounds__(32) void cs_loss_final(
    const float* __restrict__ partials,
    float*       __restrict__ out)
{
    const int lane = threadIdx.x; // 0..31, single wave

    // Each lane folds 32 block-partials; carry combined (cost - ce)
    float comb = 0.0f;
#pragma unroll
    for (int k = 0; k < BLOCKS / 32; ++k) {
        const int i = lane + 32 * k;
        comb += partials[2 * i + 1] - partials[2 * i + 0];
    }

    // A = ones(16x4), B slot0 = comb, slot1 = 0, C = 0
    v2f a; a[0] = 1.0f; a[1] = 1.0f;
    v2f b; b[0] = comb; b[1] = 0.0f;
    v8f c = {};
    c = __builtin_amdgcn_wmma_f32_16x16x4_f32(
        /*neg_a=*/false, a, /*neg_b=*/false, b,
        /*c_mod=*/(short)0, c, /*reuse_a=*/false, /*reuse_b=*/false);

    // D VGPR0: lanes 0-15 = D[0][n], lanes 16-31 = D[8][n]; both = colsum_n.
    // Butterfly over all 32 lanes -> 2 * total; halve.
    float t = c[0];
#pragma unroll
    for (int off = 16; off > 0; off >>= 1)
        t += __shfl_xor(t, off, 32);

    if (lane == 0)
        out[0] = 0.5f * t * (1.0f / (float)NROWS); // (cost_sum - ce_sum)/N
}

extern "C" void kernel_launch(void* const* d_in, const int* in_sizes, int n_in,
                              void* d_out, int out_size, void* d_ws, size_t ws_size,
                              hipStream_t stream) {
    const float* outputs = (const float*)d_in[0];
    const int*   labels  = (const int*)d_in[1];
    const float* cost    = (const float*)d_in[2];
    float*       part    = (float*)d_ws;   // 2*BLOCKS floats, fully overwritten

    cs_loss_main<<<BLOCKS, TPB, 0, stream>>>(outputs, labels, cost, part);
    cs_loss_final<<<1, 32, 0, stream>>>(part, (float*)d_out);
}